// LossLayer_27290222198842
// MI455X (gfx1250) — compile-verified
//
#include <hip/hip_runtime.h>
#include <hip/hip_bf16.h>
#include <stdint.h>

#define BATCH 512
#define NPTS  4096
#define DIMS  8
#define NCLS  40
#define KSEG  24

typedef __attribute__((ext_vector_type(2))) float v2f;
typedef __attribute__((ext_vector_type(8))) float v8f;

// ---- CDNA5 async global->LDS copy (ASYNCcnt path), inline asm so it works on
// both toolchains regardless of builtin arity. lds_off = byte offset into the
// workgroup's LDS allocation (low 32 bits of a generic pointer to __shared__).
__device__ __forceinline__ void async_copy_b128(uint32_t lds_off, const void* g) {
  asm volatile("global_load_async_to_lds_b128 %0, %1, off"
               :: "v"(lds_off), "v"(g) : "memory");
}
__device__ __forceinline__ void wait_async0() {
  asm volatile("s_wait_asynccnt 0" ::: "memory");
}

// =====================================================================
// Kernel 1: sparse-categorical cross-entropy partial sums.
// One thread per point; 40 logits = 10 float4 loads; block reduce.
// =====================================================================
__global__ __launch_bounds__(256) void ce_kernel(const float* __restrict__ sem,
                                                 const int* __restrict__ lab,
                                                 float* __restrict__ part) {
  __shared__ float red[256];
  const int t = threadIdx.x;
  const long pid = (long)blockIdx.x * 256 + t;
  const float4* row = (const float4*)(sem + pid * NCLS);
  const int lbl = lab[pid];

  float4 v[10];
#pragma unroll
  for (int i = 0; i < 10; ++i) v[i] = row[i];

  float m = -1e30f;
#pragma unroll
  for (int i = 0; i < 10; ++i)
    m = fmaxf(m, fmaxf(fmaxf(v[i].x, v[i].y), fmaxf(v[i].z, v[i].w)));

  float s = 0.f;
#pragma unroll
  for (int i = 0; i < 10; ++i)
    s += __expf(v[i].x - m) + __expf(v[i].y - m) +
         __expf(v[i].z - m) + __expf(v[i].w - m);

  const int q = lbl >> 2, r = lbl & 3;
  float xl = 0.f;
#pragma unroll
  for (int i = 0; i < 10; ++i) {
    if (i == q) {
      float4 w = v[i];
      xl = (r == 0) ? w.x : (r == 1) ? w.y : (r == 2) ? w.z : w.w;
    }
  }
  red[t] = __logf(s) + m - xl;   // -log_softmax at label
  __syncthreads();
  for (int stp = 128; stp > 0; stp >>= 1) {
    if (t < stp) red[t] += red[t + stp];
    __syncthreads();
  }
  if (t == 0) part[blockIdx.x] = red[0];
}

// =====================================================================
// Kernel 2: discriminative loss, one workgroup (8 wave32) per batch item.
// Stage 144KB to LDS via async loads; segment-sum + counts via
// V_WMMA_F32_16X16X4_F32 with a one-hot A tile (col 8 of B = 1.0 gives
// counts for free); then hinged distances from LDS.
// =====================================================================
__global__ __launch_bounds__(256) void disc_kernel(const float* __restrict__ pred,
                                                   const int* __restrict__ labels,
                                                   float* __restrict__ part) {
  __shared__ __align__(16) float s_pred[NPTS * DIMS];   // 128 KB
  __shared__ __align__(16) int   s_lab[NPTS];           //  16 KB
  __shared__ float s_acc[2][32][8];                     //   2 KB (raw WMMA C layout)
  __shared__ float s_mu[KSEG][DIMS];
  __shared__ float s_cnt[KSEG];
  __shared__ float red[256];

  const int t = threadIdx.x;
  const int b = blockIdx.x;
  const float* gp = pred + (size_t)b * NPTS * DIMS;
  const int*   gl = labels + (size_t)b * NPTS;

  const uint32_t predOff = (uint32_t)(uintptr_t)(void*)s_pred;
  const uint32_t labOff  = (uint32_t)(uintptr_t)(void*)s_lab;

  // zero WMMA merge accumulator while async copies are in flight
  if (t < 64) {
#pragma unroll
    for (int i = 0; i < 8; ++i) ((float*)s_acc)[t * 8 + i] = 0.f;
  }

  // async stage: pred = 8192 x b128, labels = 1024 x b128
  for (int c = t; c < (NPTS * DIMS * 4) / 16; c += 256)
    async_copy_b128(predOff + c * 16, (const char*)gp + c * 16);
  for (int c = t; c < (NPTS * 4) / 16; c += 256)
    async_copy_b128(labOff + c * 16, (const char*)gl + c * 16);

  wait_async0();
  __syncthreads();

  // ---- WMMA segment sums: seg_sum[K,D] = onehot[K,N] @ pred[N,D] ----
  const int lane = t & 31;
  const int wave = t >> 5;
  const int half = lane >> 4;   // A: lanes 0-15 hold K=0/1, lanes 16-31 K=2/3
  const int mrow = lane & 15;   // A row (cluster) / B column index
  const int col  = mrow & 7;    // clamped data column (branch-free load)
  const bool isdata  = (mrow < 8);
  const float onecol = (mrow == 8) ? 1.f : 0.f;  // ones column -> counts
  const int k0 = half * 2;

  v8f c0 = {};  // clusters 0..15
  v8f c1 = {};  // clusters 16..23 (+ zero padding rows)
  const int pbase = wave * 512;

  for (int ch = 0; ch < 128; ++ch) {
    const int p = pbase + ch * 4;
    const int4 lv = *(const int4*)&s_lab[p];       // labels of 4 points (broadcast)
    const int l0 = half ? lv.z : lv.x;
    const int l1 = half ? lv.w : lv.y;

    v2f a0, a1, bb;
    a0.x = (l0 == mrow)      ? 1.f : 0.f;
    a0.y = (l1 == mrow)      ? 1.f : 0.f;
    a1.x = (l0 == mrow + 16) ? 1.f : 0.f;
    a1.y = (l1 == mrow + 16) ? 1.f : 0.f;

    // unconditional LDS loads (clamped column), then lane-select: no exec
    // divergence, no saveexec around ds_load.
    const float bx = s_pred[(p + k0) * 8 + col];
    const float by = s_pred[(p + k0 + 1) * 8 + col];
    bb.x = isdata ? bx : onecol;
    bb.y = isdata ? by : onecol;

    c0 = __builtin_amdgcn_wmma_f32_16x16x4_f32(false, a0, false, bb, (short)0, c0, false, false);
    c1 = __builtin_amdgcn_wmma_f32_16x16x4_f32(false, a1, false, bb, (short)0, c1, false, false);
  }

  // merge the 8 per-wave accumulators (raw C layout) with LDS float atomics
#pragma unroll
  for (int r = 0; r < 8; ++r) {
    atomicAdd(&s_acc[0][lane][r], c0[r]);
    atomicAdd(&s_acc[1][lane][r], c1[r]);
  }
  __syncthreads();

  // C layout: element (m,n) lives at lane=(m/8)*16+n, vgpr=m%8
  if (t < KSEG) {
    const int tile = t >> 4, m = t & 15;
    s_cnt[t] = fmaxf(s_acc[tile][(m >> 3) * 16 + 8][m & 7], 1.f);
  }
  if (t < KSEG * DIMS) {
    const int k = t >> 3, d = t & 7;
    const int tile = k >> 4, m = k & 15;
    const float cnt = fmaxf(s_acc[tile][(m >> 3) * 16 + 8][m & 7], 1.f);
    s_mu[k][d] = s_acc[tile][(m >> 3) * 16 + d][m & 7] / cnt;
  }
  __syncthreads();

  // ---- l_var: sum_i hinge(L1(p_i, mu[l_i]))^2 / cnt[l_i], then /K ----
  float accv = 0.f;
  for (int i = 0; i < NPTS / 256; ++i) {
    const int p = t + i * 256;
    const int l = s_lab[p];
    const float* pp = &s_pred[p * DIMS];
    float dist = 0.f;
#pragma unroll
    for (int d = 0; d < DIMS; ++d) dist += fabsf(pp[d] - s_mu[l][d]);
    const float h = fmaxf(dist - 0.5f, 0.f);     // DELTA_V
    accv += h * h / s_cnt[l];
  }

  // ---- l_dist: off-diagonal pairwise hinge over 24x24 ----
  float accd = 0.f;
  for (int idx = t; idx < KSEG * KSEG; idx += 256) {
    const int k = idx / KSEG, j = idx - k * KSEG;
    if (k != j) {
      float md = 0.f;
#pragma unroll
      for (int d = 0; d < DIMS; ++d) md += fabsf(s_mu[k][d] - s_mu[j][d]);
      const float h = fmaxf(3.0f - md, 0.f);     // 2*DELTA_D
      accd += h * h;
    }
  }

  // ---- l_reg ----
  float accr = 0.f;
  if (t < KSEG) {
#pragma unroll
    for (int d = 0; d < DIMS; ++d) accr += fabsf(s_mu[t][d]);
  }

  red[t] = accv * (1.0f / KSEG)
         + accd * (1.0f / (KSEG * KSEG - KSEG))
         + accr * (0.001f / KSEG);
  __syncthreads();
  for (int stp = 128; stp > 0; stp >>= 1) {
    if (t < stp) red[t] += red[t + stp];
    __syncthreads();
  }
  if (t == 0) part[b] = red[0];
}

// =====================================================================
// Kernel 3: deterministic final reduction (f64 accumulate), write scalar.
// =====================================================================
__global__ __launch_bounds__(256) void final_kernel(const float* __restrict__ clsPart,
                                                    const float* __restrict__ discPart,
                                                    float* __restrict__ out) {
  __shared__ double redc[256];
  __shared__ double redd[256];
  const int t = threadIdx.x;
  double a = 0.0, d = 0.0;
  for (int i = t; i < 8192; i += 256) a += (double)clsPart[i];
  for (int i = t; i < BATCH; i += 256) d += (double)discPart[i];
  redc[t] = a; redd[t] = d;
  __syncthreads();
  for (int stp = 128; stp > 0; stp >>= 1) {
    if (t < stp) { redc[t] += redc[t + stp]; redd[t] += redd[t + stp]; }
    __syncthreads();
  }
  if (t == 0)
    out[0] = (float)(redc[0] / ((double)BATCH * NPTS) + redd[0] / (double)BATCH);
}

extern "C" void kernel_launch(void* const* d_in, const int* in_sizes, int n_in,
                              void* d_out, int out_size, void* d_ws, size_t ws_size,
                              hipStream_t stream) {
  (void)in_sizes; (void)n_in; (void)out_size; (void)ws_size;
  const float* pred_ins = (const float*)d_in[0];   // [512,4096,8]
  const float* pred_sem = (const float*)d_in[1];   // [512,4096,40]
  const int*   true_ins = (const int*)d_in[2];     // [512,4096]
  const int*   true_sem = (const int*)d_in[3];     // [512,4096]

  float* ws = (float*)d_ws;
  float* clsPart  = ws;          // 8192 floats
  float* discPart = ws + 8192;   // 512 floats

  ce_kernel<<<8192, 256, 0, stream>>>(pred_sem, true_sem, clsPart);
  disc_kernel<<<BATCH, 256, 0, stream>>>(pred_ins, true_ins, discPart);
  final_kernel<<<1, 256, 0, stream>>>(clsPart, discPart, (float*)d_out);
}